// HyperGNN_81681688035839
// MI455X (gfx1250) — compile-verified
//
#include <hip/hip_runtime.h>
#include <hip/hip_bf16.h>

typedef __attribute__((ext_vector_type(2))) float v2f;
typedef __attribute__((ext_vector_type(8))) float v8f;

// ---------------------------------------------------------------------------
// Per-layer init: acc <- bias, m <- -inf, z <- 0, ma <- 0
// ---------------------------------------------------------------------------
__global__ void gat_init(float* __restrict__ acc, const float* __restrict__ bias,
                         float* __restrict__ m, float* __restrict__ z,
                         float* __restrict__ ma, int n) {
    int t = blockIdx.x * blockDim.x + threadIdx.x;
    if (t < 64) ma[t] = 0.0f;
    if (t >= n * 32) return;
    acc[t] = bias[t];
    if ((t & 31) == 0) {
        int node = t >> 5;
        m[node] = -3.0e38f;
        z[node] = 0.0f;
    }
}

// ---------------------------------------------------------------------------
// mean_att accumulation: ma[k] = sum_n a[n*64+k]   (scaled by 1/N later)
// launched with 128 blocks x 256 threads (32768 threads, 512 per column)
// ---------------------------------------------------------------------------
__global__ void gat_mean_att(const float* __restrict__ a, float* __restrict__ ma, int n) {
    int tid = blockIdx.x * blockDim.x + threadIdx.x;
    int k = tid & 63;
    int stride = (gridDim.x * blockDim.x) >> 6;
    float s = 0.0f;
    for (int row = tid >> 6; row < n; row += stride)
        s += a[(size_t)row * 64 + k];
    atomicAdd(&ma[k], s);
}

// ---------------------------------------------------------------------------
// Per-node bmm via V_WMMA_F32_16X16X4_F32: h[n] = x[n] @ w[n]
// One wave per node. A = x replicated over 16 rows (16x4 f32 per step),
// B = 4x16 column-slab of w[n]; 8 K-chunks x 2 column halves = 16 WMMAs.
// A 16x4 f32 layout: v0: K=0 (lanes 0-15) / K=2 (lanes 16-31); v1: K=1 / K=3.
// B  4x16 f32 layout: v0: K=0 row (lanes 0-15) / K=2 row (lanes 16-31); v1: K=1/K=3.
// D VGPR0: lanes 0-15 -> N=lane (cols 0-15); lanes 16-31 of half-2 -> cols 16-31.
// ---------------------------------------------------------------------------
__global__ void gat_bmm_wmma(const float* __restrict__ x, const float* __restrict__ w,
                             float* __restrict__ h, int n) {
    int wave = (int)((blockIdx.x * blockDim.x + threadIdx.x) >> 5);
    int lane = threadIdx.x & 31;
    if (wave >= n) return;                      // uniform per wave: EXEC stays all-1s
    const float* __restrict__ xr = x + (size_t)wave * 32;
    const float* __restrict__ wr = w + (size_t)wave * 1024;
    const int hi = lane >> 4;                   // half-wave selector
    const int ln = lane & 15;
    v8f c0 = {};                                // columns 0..15
    v8f c1 = {};                                // columns 16..31
#pragma unroll
    for (int t = 0; t < 8; ++t) {
        const int k0 = 4 * t + 2 * hi;          // K rows this half-wave supplies
        v2f av;  av[0] = xr[k0];                av[1] = xr[k0 + 1];
        v2f b0v; b0v[0] = wr[k0 * 32 + ln];     b0v[1] = wr[(k0 + 1) * 32 + ln];
        v2f b1v; b1v[0] = wr[k0 * 32 + 16 + ln]; b1v[1] = wr[(k0 + 1) * 32 + 16 + ln];
        c0 = __builtin_amdgcn_wmma_f32_16x16x4_f32(false, av, false, b0v,
                                                   (short)0, c0, false, false);
        c1 = __builtin_amdgcn_wmma_f32_16x16x4_f32(false, av, false, b1v,
                                                   (short)0, c1, false, false);
    }
    // lane<16: c0 VGPR0 holds (M=0, N=lane) = h[lane]
    // lane>=16: c1 VGPR0 holds (M=8, N=lane-16) = h[16 + lane-16] = h[lane]
    h[(size_t)wave * 32 + lane] = hi ? c1[0] : c0[0];
}

// ---------------------------------------------------------------------------
// Per-node attention scalars: p[n] = h[n] . ma[0:32]/N, q[n] = h[n] . ma[32:64]/N
// ---------------------------------------------------------------------------
__global__ void gat_node_pq(const float* __restrict__ h, const float* __restrict__ ma,
                            float* __restrict__ p, float* __restrict__ q,
                            int n, float invN) {
    int wave = (int)((blockIdx.x * blockDim.x + threadIdx.x) >> 5);
    int lane = threadIdx.x & 31;
    if (wave >= n) return;
    float hv = h[(size_t)wave * 32 + lane];
    float pa = hv * ma[lane] * invN;
    float qa = hv * ma[32 + lane] * invN;
#pragma unroll
    for (int off = 16; off; off >>= 1) {
        pa += __shfl_xor(pa, off, 32);
        qa += __shfl_xor(qa, off, 32);
    }
    if (lane == 0) { p[wave] = pa; q[wave] = qa; }
}

__device__ __forceinline__ float leaky02(float s) { return s > 0.0f ? s : 0.2f * s; }

// ---------------------------------------------------------------------------
// Edge pass 1: segment max  (lowers to GLOBAL_ATOMIC_MAX_NUM_F32)
// ---------------------------------------------------------------------------
__global__ void gat_edge_max(const int* __restrict__ ei, const float* __restrict__ p,
                             const float* __restrict__ q, float* __restrict__ m, int e) {
    int t = blockIdx.x * blockDim.x + threadIdx.x;
    if (t >= e) return;
    int r = ei[t], c = ei[e + t];
    float s = leaky02(p[r] + q[c]);
    atomicMax(&m[c], s);
}

// ---------------------------------------------------------------------------
// Edge pass 2: z[c] += exp(s - m[c])
// ---------------------------------------------------------------------------
__global__ void gat_edge_expsum(const int* __restrict__ ei, const float* __restrict__ p,
                                const float* __restrict__ q, const float* __restrict__ m,
                                float* __restrict__ z, int e) {
    int t = blockIdx.x * blockDim.x + threadIdx.x;
    if (t >= e) return;
    int r = ei[t], c = ei[e + t];
    float s = leaky02(p[r] + q[c]);
    atomicAdd(&z[c], __expf(s - m[c]));
}

// ---------------------------------------------------------------------------
// Edge pass 3: acc[c] += h[r] * att  (wave per edge; h/acc tables are L2-resident)
// ---------------------------------------------------------------------------
__global__ void gat_edge_agg(const int* __restrict__ ei, const float* __restrict__ p,
                             const float* __restrict__ q, const float* __restrict__ m,
                             const float* __restrict__ z, const float* __restrict__ h,
                             float* __restrict__ acc, int e) {
    int wave = (int)((blockIdx.x * blockDim.x + threadIdx.x) >> 5);
    int lane = threadIdx.x & 31;
    if (wave >= e) return;
    int r = ei[wave], c = ei[e + wave];
    float s = leaky02(p[r] + q[c]);
    float att = __expf(s - m[c]) / z[c];
    atomicAdd(&acc[(size_t)c * 32 + lane], h[(size_t)r * 32 + lane] * att);
}

// ---------------------------------------------------------------------------
// In-place ReLU (inter-layer)
// ---------------------------------------------------------------------------
__global__ void gat_relu(float* __restrict__ buf, int count) {
    int t = blockIdx.x * blockDim.x + threadIdx.x;
    if (t >= count) return;
    float v = buf[t];
    buf[t] = v > 0.0f ? v : 0.0f;
}

extern "C" void kernel_launch(void* const* d_in, const int* in_sizes, int n_in,
                              void* d_out, int out_size, void* d_ws, size_t ws_size,
                              hipStream_t stream) {
    const float* x  = (const float*)d_in[0];
    const int*   ei = (const int*)  d_in[1];
    const float* w0 = (const float*)d_in[2];
    const float* a0 = (const float*)d_in[3];
    const float* b0 = (const float*)d_in[4];
    const float* w1 = (const float*)d_in[5];
    const float* a1 = (const float*)d_in[6];
    const float* b1 = (const float*)d_in[7];
    float* out = (float*)d_out;

    const int N = in_sizes[0] / 32;
    const int E = in_sizes[1] / 2;
    const float invN = 1.0f / (float)N;

    // Workspace layout (floats): ma[64] | p[N] | q[N] | m[N] | z[N] | h[N*32] | acc[N*32]
    float* ws  = (float*)d_ws;
    float* ma  = ws;
    float* p   = ma + 64;
    float* q   = p + N;
    float* m   = q + N;
    float* z   = m + N;
    float* h   = z + N;
    float* acc = h + (size_t)N * 32;

    const int B = 256;
    dim3 gNode32((N * 32 + B - 1) / B);   // thread per (node, channel)
    dim3 gNodeW((N + 7) / 8);             // wave per node (8 waves / block)
    dim3 gEdge((E + B - 1) / B);          // thread per edge
    dim3 gEdgeW((E + 7) / 8);             // wave per edge

    // -------- Layer 1: acc = segsoftmax-agg(x @ w0) + b0; x2 = relu(acc) --------
    gat_init      <<<gNode32, B, 0, stream>>>(acc, b0, m, z, ma, N);
    gat_mean_att  <<<128,     B, 0, stream>>>(a0, ma, N);
    gat_bmm_wmma  <<<gNodeW,  B, 0, stream>>>(x, w0, h, N);
    gat_node_pq   <<<gNodeW,  B, 0, stream>>>(h, ma, p, q, N, invN);
    gat_edge_max  <<<gEdge,   B, 0, stream>>>(ei, p, q, m, E);
    gat_edge_expsum<<<gEdge,  B, 0, stream>>>(ei, p, q, m, z, E);
    gat_edge_agg  <<<gEdgeW,  B, 0, stream>>>(ei, p, q, m, z, h, acc, E);
    gat_relu      <<<gNode32, B, 0, stream>>>(acc, N * 32);

    // -------- Layer 2: out = segsoftmax-agg(acc @ w1) + b1 --------
    gat_init      <<<gNode32, B, 0, stream>>>(out, b1, m, z, ma, N);
    gat_mean_att  <<<128,     B, 0, stream>>>(a1, ma, N);
    gat_bmm_wmma  <<<gNodeW,  B, 0, stream>>>(acc, w1, h, N);
    gat_node_pq   <<<gNodeW,  B, 0, stream>>>(h, ma, p, q, N, invN);
    gat_edge_max  <<<gEdge,   B, 0, stream>>>(ei, p, q, m, E);
    gat_edge_expsum<<<gEdge,  B, 0, stream>>>(ei, p, q, m, z, E);
    gat_edge_agg  <<<gEdgeW,  B, 0, stream>>>(ei, p, q, m, z, h, out, E);
}